// QuantumRegression_65481071407732
// MI455X (gfx1250) — compile-verified
//
#include <hip/hip_runtime.h>
#include <hip/hip_bf16.h>

#define NW     14
#define DIM    (1 << NW)      // 16384 amplitudes
#define TPB    256            // 8 wave32 per workgroup
#define NLAYER 3

typedef float v2f __attribute__((ext_vector_type(2)));
typedef float v8f __attribute__((ext_vector_type(8)));

struct C2 { float x, y; };
__device__ __forceinline__ C2 cmk(float a, float b) { C2 r; r.x = a; r.y = b; return r; }
__device__ __forceinline__ C2 cmul(C2 a, C2 b) {
    return cmk(a.x * b.x - a.y * b.y, a.x * b.y + a.y * b.x);
}
__device__ __forceinline__ C2 cadd(C2 a, C2 b) { return cmk(a.x + b.x, a.y + b.y); }

// Apply a generic complex 2x2 gate on wire w (wire 0 = MSB of the basis index).
__device__ __forceinline__ void apply1q(C2* psi, int w, C2 g00, C2 g01, C2 g10, C2 g11,
                                        int tid) {
    const int s = NW - 1 - w;       // bit position of this wire
    const int M = 1 << s;
    for (int p = tid; p < DIM / 2; p += TPB) {
        const int i0 = ((p >> s) << (s + 1)) | (p & (M - 1));
        const int i1 = i0 | M;
        const C2 a = psi[i0];
        const C2 b = psi[i1];
        psi[i0] = cadd(cmul(g00, a), cmul(g01, b));
        psi[i1] = cadd(cmul(g10, a), cmul(g11, b));
    }
    __syncthreads();
}

// One workgroup per batch sample: full 128KB state vector lives in LDS.
__global__ __launch_bounds__(TPB) void qsim_kernel(const float* __restrict__ state_batch,
                                                   const float* __restrict__ var_params,
                                                   float* __restrict__ zout /* [B][16] */) {
    __shared__ C2 psi[DIM];         // 128 KB of the WGP's 320 KB LDS
    __shared__ float zsum[16];

    const int tid = threadIdx.x;
    const int b   = blockIdx.x;

    // |0...0>
    for (int i = tid; i < DIM; i += TPB) psi[i] = cmk(i == 0 ? 1.f : 0.f, 0.f);
    if (tid < 16) zsum[tid] = 0.f;
    __syncthreads();

    // --- RX encoding with per-sample angles: [[c, -i s], [-i s, c]] ---
    for (int w = 0; w < NW; ++w) {
        const float h = 0.5f * state_batch[b * NW + w];
        const float c = cosf(h), s = sinf(h);
        apply1q(psi, w, cmk(c, 0.f), cmk(0.f, -s), cmk(0.f, -s), cmk(c, 0.f), tid);
    }

    // --- variational layers: fused RZ(th) * RY(a) per wire, then CNOT chain ---
    for (int l = 0; l < NLAYER; ++l) {
        for (int w = 0; w < NW; ++w) {
            const float a  = var_params[(l * NW + w) * 3 + 0];
            const float th = var_params[(l * NW + w) * 3 + 1];
            const float ca = cosf(0.5f * a),  sa = sinf(0.5f * a);
            const float ct = cosf(0.5f * th), st = sinf(0.5f * th);
            // G = diag(e^{-i th/2}, e^{+i th/2}) * [[ca, -sa], [sa, ca]]
            const C2 g00 = cmk( ca * ct, -ca * st);
            const C2 g01 = cmk(-sa * ct,  sa * st);
            const C2 g10 = cmk( sa * ct,  sa * st);
            const C2 g11 = cmk( ca * ct,  ca * st);
            apply1q(psi, w, g00, g01, g10, g11, tid);
        }
        // CNOT(c, c+1) chain: swap target bit where control bit == 1 (in place)
        for (int c = 0; c < NW - 1; ++c) {
            const int s = NW - 2 - c;   // target bit position; control = s+1
            for (int q = tid; q < DIM / 4; q += TPB) {
                const int low = q & ((1 << s) - 1);
                const int i0  = ((q >> s) << (s + 2)) | (2 << s) | low; // ctrl=1, tgt=0
                const int i1  = i0 | (1 << s);
                const C2 t = psi[i0];
                psi[i0] = psi[i1];
                psi[i1] = t;
            }
            __syncthreads();
        }
    }

    // --- <Z_w> = sum_i (+/-) |psi_i|^2, sign from bit (NW-1-w) of i ---
    float acc[NW];
#pragma unroll
    for (int w = 0; w < NW; ++w) acc[w] = 0.f;
    for (int i = tid; i < DIM; i += TPB) {
        const C2 v = psi[i];
        const float p = v.x * v.x + v.y * v.y;
#pragma unroll
        for (int w = 0; w < NW; ++w)
            acc[w] += ((i >> (NW - 1 - w)) & 1) ? -p : p;
    }
#pragma unroll
    for (int w = 0; w < NW; ++w) atomicAdd(&zsum[w], acc[w]);   // ds_add_f32
    __syncthreads();

    if (tid < 16) zout[b * 16 + tid] = (tid < NW) ? zsum[tid] : 0.f;  // pad cols 14,15
}

// Linear head out[b] = sum_w z[b][w]*hw[w] + hb via V_WMMA_F32_16X16X4_F32.
// One wave per block handles a 16-row tile; head weights are broadcast across
// all N columns of B, so only the K mapping of the B layout matters; result is
// read from column N=0 (lanes 0 and 16 per the f32 C/D layout).
__global__ __launch_bounds__(32) void head_wmma_kernel(const float* __restrict__ z,
                                                       const float* __restrict__ hw,
                                                       const float* __restrict__ hb,
                                                       float* __restrict__ out) {
    const int lane = threadIdx.x;           // 0..31, wave32
    const int m    = lane & 15;             // A-matrix row (M)
    const int kb   = (lane >> 4) << 1;      // K sub-offset: lanes 0-15 -> 0, 16-31 -> 2
    const int row  = blockIdx.x * 16 + m;

    v8f c = {0.f, 0.f, 0.f, 0.f, 0.f, 0.f, 0.f, 0.f};
    for (int kc = 0; kc < 16; kc += 4) {
        v2f a, bb;
        a.x  = z[row * 16 + kc + kb];       // padded cols 14,15 are zero
        a.y  = z[row * 16 + kc + kb + 1];
        bb.x = (kc + kb     < NW) ? hw[kc + kb]     : 0.f;
        bb.y = (kc + kb + 1 < NW) ? hw[kc + kb + 1] : 0.f;
        c = __builtin_amdgcn_wmma_f32_16x16x4_f32(
                /*neg_a=*/false, a, /*neg_b=*/false, bb,
                /*c_mod=*/(short)0, c, /*reuse_a=*/false, /*reuse_b=*/false);
    }
    // D layout: VGPR r -> M=r (lanes 0-15) / M=r+8 (lanes 16-31), N = lane&15.
    if ((lane & 15) == 0) {
        const int mbase = (lane >> 4) * 8;
        const float bias = hb[0];
#pragma unroll
        for (int r = 0; r < 8; ++r)
            out[blockIdx.x * 16 + mbase + r] = c[r] + bias;
    }
}

extern "C" void kernel_launch(void* const* d_in, const int* in_sizes, int n_in,
                              void* d_out, int out_size, void* d_ws, size_t ws_size,
                              hipStream_t stream) {
    const float* state_batch = (const float*)d_in[0];  // [B, 14] f32
    const float* var_params  = (const float*)d_in[1];  // [3, 14, 3] f32
    const float* head_w      = (const float*)d_in[2];  // [1, 14] f32
    const float* head_b      = (const float*)d_in[3];  // [1] f32
    float*       out         = (float*)d_out;          // [B] f32
    float*       z_pad       = (float*)d_ws;           // [B, 16] f32 scratch

    const int B = in_sizes[0] / NW;                    // 256

    qsim_kernel<<<B, TPB, 0, stream>>>(state_batch, var_params, z_pad);
    head_wmma_kernel<<<B / 16, 32, 0, stream>>>(z_pad, head_w, head_b, out);
}